// ConnectedComponentsModule_14645838479605
// MI455X (gfx1250) — compile-verified
//
#include <hip/hip_runtime.h>
#include <hip/hip_bf16.h>

typedef __attribute__((ext_vector_type(2))) float v2f;
typedef __attribute__((ext_vector_type(8))) float v8f;

namespace {
constexpr int B_     = 512;
constexpr int NCAT   = 10;
constexpr int GRID_  = 64;
constexpr int H_     = 64;
constexpr int W_     = 64;
constexpr int HW_    = H_ * W_;
constexpr int MAXOBJ = 20;
constexpr int FEATD  = 128;
constexpr int NOBJ   = B_ * MAXOBJ;           // 10240
constexpr int FEATS_N = NOBJ * FEATD;         // 1,310,720
}

__device__ __forceinline__ int imin_(int a, int b) { return a < b ? a : b; }
__device__ __forceinline__ int imax_(int a, int b) { return a > b ? a : b; }

// ---------------------------------------------------------------------------
// Kernel 1: per-pixel argmax over NCAT channels -> color [B,H,W]
// ---------------------------------------------------------------------------
__global__ void argmax_color_kernel(const float* __restrict__ x, int* __restrict__ color) {
    int idx = blockIdx.x * blockDim.x + threadIdx.x;
    if (idx >= B_ * HW_) return;
    int b = idx / HW_;
    int p = idx - b * HW_;
    const float* px = x + (size_t)b * NCAT * HW_ + p;
    float best = px[0];
    int bi = 0;
#pragma unroll
    for (int c = 1; c < NCAT; ++c) {
        float v = px[(size_t)c * HW_];
        if (v > best) { best = v; bi = c; }   // first-max tie rule like jnp.argmax
    }
    color[idx] = bi;
}

// ---------------------------------------------------------------------------
// Kernel 2: connected components (8-conn, same color, bg=0), one WG per sample.
// LDS-resident min-label propagation with pointer jumping. Converges to
// per-component minimum flat index == reference fixed point. INF = HW.
// ---------------------------------------------------------------------------
__global__ void ccl_kernel(const int* __restrict__ colorG, int* __restrict__ labG) {
    __shared__ int lab[HW_];
    __shared__ int col[HW_];
    __shared__ int changed;
    const int b = blockIdx.x;
    const int tid = threadIdx.x;
    const int* c = colorG + (size_t)b * HW_;

    for (int p = tid; p < HW_; p += 256) {
        int cc = c[p];
        col[p] = cc;
        lab[p] = (cc > 0) ? p : HW_;
    }
    __syncthreads();

    for (int it = 0; it < 192; ++it) {
        if (tid == 0) changed = 0;
        __syncthreads();

        int nl[16];
        int lc = 0;
#pragma unroll
        for (int i = 0; i < 16; ++i) {
            int p = tid + i * 256;
            int l = lab[p];
            int m = l;
            if (l < HW_) {
                int cc = col[p];
                int r = p >> 6, q = p & 63;
                // 8 neighbors, bounds-checked, same-color only
                if (r > 0) {
                    if (q > 0)  { int np = p - 65; if (col[np] == cc) m = imin_(m, lab[np]); }
                    { int np = p - 64; if (col[np] == cc) m = imin_(m, lab[np]); }
                    if (q < 63) { int np = p - 63; if (col[np] == cc) m = imin_(m, lab[np]); }
                }
                if (q > 0)  { int np = p - 1; if (col[np] == cc) m = imin_(m, lab[np]); }
                if (q < 63) { int np = p + 1; if (col[np] == cc) m = imin_(m, lab[np]); }
                if (r < 63) {
                    if (q > 0)  { int np = p + 63; if (col[np] == cc) m = imin_(m, lab[np]); }
                    { int np = p + 64; if (col[np] == cc) m = imin_(m, lab[np]); }
                    if (q < 63) { int np = p + 65; if (col[np] == cc) m = imin_(m, lab[np]); }
                }
                // pointer jump (m <= l < HW, so lab[m] is a valid read)
                m = imin_(m, lab[m]);
                if (m != l) lc = 1;
            }
            nl[i] = m;
        }
        __syncthreads();
#pragma unroll
        for (int i = 0; i < 16; ++i) lab[tid + i * 256] = nl[i];
        if (lc) changed = 1;
        __syncthreads();
        if (!changed) break;
    }

    int* out = labG + (size_t)b * HW_;
    for (int p = tid; p < HW_; p += 256) out[p] = lab[p];
}

// ---------------------------------------------------------------------------
// Kernel 3: per-sample object extraction: segment sizes, top-20 (max value,
// min-index tie break), bbox, osize, obj color, vobj, masks.
// ---------------------------------------------------------------------------
__global__ void extract_kernel(const int* __restrict__ labG, const int* __restrict__ colorG,
                               float* __restrict__ bboxW, float* __restrict__ osizeW,
                               int* __restrict__ ocolW,
                               float* __restrict__ vobjOut, float* __restrict__ maskOut) {
    __shared__ int labS[HW_];
    __shared__ int sz[HW_];
    __shared__ int redv[256];
    __shared__ int redi[256];
    __shared__ int topId[MAXOBJ];
    __shared__ int topVal[MAXOBJ];
    __shared__ int rmn[MAXOBJ], rmx[MAXOBJ], cmn[MAXOBJ], cmx[MAXOBJ];

    const int b = blockIdx.x;
    const int tid = threadIdx.x;

    for (int p = tid; p < HW_; p += 256) {
        labS[p] = labG[(size_t)b * HW_ + p];
        sz[p] = 0;
    }
    __syncthreads();
    for (int p = tid; p < HW_; p += 256) {
        int l = labS[p];
        if (l < HW_) atomicAdd(&sz[l], 1);
    }
    __syncthreads();

    // 20 sequential arg-max passes; score = (size>=2 ? size : 0), picked -> -1
    for (int k = 0; k < MAXOBJ; ++k) {
        int bv = -2, biv = HW_;
        for (int p = tid; p < HW_; p += 256) {
            int s = sz[p];
            int sc = (s >= 2) ? s : (s < 0 ? -1 : 0);
            if (sc > bv) { bv = sc; biv = p; }    // p ascending -> min index per thread
        }
        redv[tid] = bv; redi[tid] = biv;
        __syncthreads();
        for (int s2 = 128; s2 > 0; s2 >>= 1) {
            if (tid < s2) {
                int v2 = redv[tid + s2], i2 = redi[tid + s2];
                if (v2 > redv[tid] || (v2 == redv[tid] && i2 < redi[tid])) {
                    redv[tid] = v2; redi[tid] = i2;
                }
            }
            __syncthreads();
        }
        if (tid == 0) {
            topId[k] = redi[0];
            topVal[k] = redv[0];
            sz[redi[0]] = -1;      // mark picked
        }
        __syncthreads();
    }

    if (tid < MAXOBJ) { rmn[tid] = H_; rmx[tid] = -1; cmn[tid] = W_; cmx[tid] = -1; }
    __syncthreads();

    for (int p = tid; p < HW_; p += 256) {
        int l = labS[p];
        if (l >= HW_) continue;
        int r = p >> 6, q = p & 63;
#pragma unroll
        for (int k = 0; k < MAXOBJ; ++k) {
            if (topId[k] == l) {
                atomicMin(&rmn[k], r); atomicMax(&rmx[k], r);
                atomicMin(&cmn[k], q); atomicMax(&cmx[k], q);
            }
        }
    }
    __syncthreads();

    if (tid < MAXOBJ) {
        int k = tid;
        bool vb = topVal[k] >= 2;
        int g = b * MAXOBJ + k;
        vobjOut[g] = vb ? 1.0f : 0.0f;
        ocolW[g] = vb ? colorG[(size_t)b * HW_ + topId[k]] : 0;
        osizeW[g] = vb ? (float)topVal[k] / (float)(GRID_ * GRID_) : 0.0f;
        float s = 1.0f / (float)GRID_;
        bboxW[g * 4 + 0] = vb ? cmn[k] * s : 0.0f;
        bboxW[g * 4 + 1] = vb ? rmn[k] * s : 0.0f;
        bboxW[g * 4 + 2] = vb ? cmx[k] * s : 0.0f;
        bboxW[g * 4 + 3] = vb ? rmx[k] * s : 0.0f;
    }
    __syncthreads();

    for (int k = 0; k < MAXOBJ; ++k) {
        int idt = topId[k];
        bool vb = topVal[k] >= 2;
        float* mrow = maskOut + ((size_t)(b * MAXOBJ + k)) * HW_;
        for (int p = tid; p < HW_; p += 256)
            mrow[p] = (vb && labS[p] == idt) ? 1.0f : 0.0f;
    }
}

// ---------------------------------------------------------------------------
// Kernel 4: encoder. 16 objects per block, 256 threads (8 waves).
// Phase 1: small MLPs -> comb[16][128] in LDS.
// Phase 2: comb @ comb_w via V_WMMA_F32_16X16X4_F32, one 16x16 tile per wave.
// Phase 3: row LayerNorm + ReLU + vobj mask -> feats.
// ---------------------------------------------------------------------------
__global__ void encode_kernel(const float* __restrict__ w1, const float* __restrict__ b1,
                              const float* __restrict__ g1, const float* __restrict__ be1,
                              const float* __restrict__ w2, const float* __restrict__ b2,
                              const float* __restrict__ g2, const float* __restrict__ be2,
                              const float* __restrict__ cemb,
                              const float* __restrict__ sw, const float* __restrict__ sb,
                              const float* __restrict__ sg, const float* __restrict__ sbe,
                              const float* __restrict__ cw, const float* __restrict__ cb,
                              const float* __restrict__ cg, const float* __restrict__ cbe,
                              const float* __restrict__ bboxW, const float* __restrict__ osizeW,
                              const int* __restrict__ ocolW, const float* __restrict__ vobjOut,
                              float* __restrict__ feats) {
    __shared__ float combS[16][FEATD];
    __shared__ float accS[16][FEATD];
    __shared__ float vflag[16];

    const int tid = threadIdx.x;
    const int objBase = blockIdx.x * 16;

    if (tid < 16) {
        const int g = objBase + tid;
        float bx[4];
#pragma unroll
        for (int i = 0; i < 4; ++i) bx[i] = bboxW[g * 4 + i];
        const float os = osizeW[g];
        const int cid = ocolW[g];
        vflag[tid] = vobjOut[g];

        // h1 = relu(LN(bbox @ w1 + b1))
        float h1[32];
        float mn = 0.f;
#pragma unroll
        for (int j = 0; j < 32; ++j) {
            float s = b1[j];
#pragma unroll
            for (int i = 0; i < 4; ++i) s += bx[i] * w1[i * 32 + j];
            h1[j] = s; mn += s;
        }
        mn *= (1.f / 32.f);
        float vr = 0.f;
#pragma unroll
        for (int j = 0; j < 32; ++j) { float d = h1[j] - mn; vr += d * d; }
        vr *= (1.f / 32.f);
        float inv = rsqrtf(vr + 1e-5f);
#pragma unroll
        for (int j = 0; j < 32; ++j) {
            float y = (h1[j] - mn) * inv * g1[j] + be1[j];
            h1[j] = y > 0.f ? y : 0.f;
        }

        // h2 = relu(LN(h1 @ w2 + b2))
        float h2[64];
        mn = 0.f;
#pragma unroll
        for (int j = 0; j < 64; ++j) {
            float s = b2[j];
#pragma unroll
            for (int i = 0; i < 32; ++i) s += h1[i] * w2[i * 64 + j];
            h2[j] = s; mn += s;
        }
        mn *= (1.f / 64.f);
        vr = 0.f;
#pragma unroll
        for (int j = 0; j < 64; ++j) { float d = h2[j] - mn; vr += d * d; }
        vr *= (1.f / 64.f);
        inv = rsqrtf(vr + 1e-5f);
#pragma unroll
        for (int j = 0; j < 64; ++j) {
            float y = (h2[j] - mn) * inv * g2[j] + be2[j];
            combS[tid][j] = y > 0.f ? y : 0.f;
        }

        // color embedding
#pragma unroll
        for (int j = 0; j < 32; ++j) combS[tid][64 + j] = cemb[cid * 32 + j];

        // se = relu(LN(osize * size_w + size_b))
        float se[32];
        mn = 0.f;
#pragma unroll
        for (int j = 0; j < 32; ++j) { float s = sb[j] + os * sw[j]; se[j] = s; mn += s; }
        mn *= (1.f / 32.f);
        vr = 0.f;
#pragma unroll
        for (int j = 0; j < 32; ++j) { float d = se[j] - mn; vr += d * d; }
        vr *= (1.f / 32.f);
        inv = rsqrtf(vr + 1e-5f);
#pragma unroll
        for (int j = 0; j < 32; ++j) {
            float y = (se[j] - mn) * inv * sg[j] + sbe[j];
            combS[tid][96 + j] = y > 0.f ? y : 0.f;
        }
    }
    __syncthreads();

    // ---- WMMA GEMM: each wave computes 16 rows x 16 cols, K=128 via 32 steps ----
    const int wave = tid >> 5;
    const int lane = tid & 31;
    const int n0 = wave * 16;
    const int mrow = lane & 15;
    const int ksel = (lane < 16) ? 0 : 2;

    v8f acc = {};
#pragma unroll
    for (int kb = 0; kb < FEATD; kb += 4) {
        v2f a, bf;
        a.x = combS[mrow][kb + ksel];
        a.y = combS[mrow][kb + ksel + 1];
        bf.x = cw[(kb + ksel) * FEATD + n0 + mrow];
        bf.y = cw[(kb + ksel + 1) * FEATD + n0 + mrow];
        acc = __builtin_amdgcn_wmma_f32_16x16x4_f32(
            /*neg_a=*/false, a, /*neg_b=*/false, bf,
            /*c_mod=*/(short)0, acc, /*reuse_a=*/false, /*reuse_b=*/false);
    }
#pragma unroll
    for (int i = 0; i < 8; ++i) {
        int row = (lane < 16) ? i : (i + 8);
        accS[row][n0 + mrow] = acc[i];
    }
    __syncthreads();

    // ---- LayerNorm over 128 + ReLU + vobj mask ----
    if (tid < 16) {
        float mn = 0.f;
#pragma unroll
        for (int c = 0; c < FEATD; ++c) mn += accS[tid][c] + cb[c];
        mn *= (1.f / 128.f);
        float vr = 0.f;
#pragma unroll
        for (int c = 0; c < FEATD; ++c) { float d = (accS[tid][c] + cb[c]) - mn; vr += d * d; }
        vr *= (1.f / 128.f);
        float inv = rsqrtf(vr + 1e-5f);
        float vf = vflag[tid];
        float* outp = feats + (size_t)(objBase + tid) * FEATD;
#pragma unroll
        for (int c = 0; c < FEATD; ++c) {
            float v = accS[tid][c] + cb[c];
            float y = (v - mn) * inv * cg[c] + cbe[c];
            y = y > 0.f ? y : 0.f;
            outp[c] = y * vf;
        }
    }
}

// ---------------------------------------------------------------------------
extern "C" void kernel_launch(void* const* d_in, const int* in_sizes, int n_in,
                              void* d_out, int out_size, void* d_ws, size_t ws_size,
                              hipStream_t stream) {
    const float* x   = (const float*)d_in[0];
    const float* w1  = (const float*)d_in[1];
    const float* b1  = (const float*)d_in[2];
    const float* g1  = (const float*)d_in[3];
    const float* be1 = (const float*)d_in[4];
    const float* w2  = (const float*)d_in[5];
    const float* b2  = (const float*)d_in[6];
    const float* g2  = (const float*)d_in[7];
    const float* be2 = (const float*)d_in[8];
    const float* ce  = (const float*)d_in[9];
    const float* sw  = (const float*)d_in[10];
    const float* sb  = (const float*)d_in[11];
    const float* sg  = (const float*)d_in[12];
    const float* sbe = (const float*)d_in[13];
    const float* cw  = (const float*)d_in[14];
    const float* cb  = (const float*)d_in[15];
    const float* cg  = (const float*)d_in[16];
    const float* cbe = (const float*)d_in[17];

    // workspace layout
    int*   colorW = (int*)d_ws;                     // B*HW
    int*   labW   = colorW + (size_t)B_ * HW_;      // B*HW
    float* bboxW  = (float*)(labW + (size_t)B_ * HW_);  // NOBJ*4
    float* osizeW = bboxW + (size_t)NOBJ * 4;       // NOBJ
    int*   ocolW  = (int*)(osizeW + NOBJ);          // NOBJ

    // output layout: feats [B,20,128] | vobj [B,20] | masks [B,20,H,W]
    float* feats  = (float*)d_out;
    float* vobjO  = feats + FEATS_N;
    float* maskO  = vobjO + NOBJ;

    const int totalPix = B_ * HW_;
    argmax_color_kernel<<<(totalPix + 255) / 256, 256, 0, stream>>>(x, colorW);
    ccl_kernel<<<B_, 256, 0, stream>>>(colorW, labW);
    extract_kernel<<<B_, 256, 0, stream>>>(labW, colorW, bboxW, osizeW, ocolW, vobjO, maskO);
    encode_kernel<<<NOBJ / 16, 256, 0, stream>>>(w1, b1, g1, be1, w2, b2, g2, be2,
                                                 ce, sw, sb, sg, sbe, cw, cb, cg, cbe,
                                                 bboxW, osizeW, ocolW, vobjO, feats);
}